// LieSE3_66649302499845
// MI455X (gfx1250) — compile-verified
//
#include <hip/hip_runtime.h>
#include <hip/hip_bf16.h>

// SE(2) exponential map, batched: uv[B,3] (x, y, yaw) -> out[B,4,4].
// Pure streaming / elementwise op: ~304 MiB moved, ~0.1 GFLOP.
// Memory-bound on MI455X (23.3 TB/s -> ~13.4 us floor). WMMA intentionally
// NOT used: no shared contraction dimension exists; tensor ops cannot reduce
// bytes or flops here. Strategy: 128-bit NT loads/stores, fast VALU
// transcendentals, persistent grid with global_prefetch_b8 one tile ahead.

typedef __attribute__((ext_vector_type(4))) float v4f;

#define ROTATION_SCALE 0.1f
#define LIE_EPS 1e-5f

__device__ __forceinline__ void se2_rows(float x, float y, float p,
                                         v4f& row0, v4f& row1) {
    const float w     = p * ROTATION_SCALE;
    const float t2    = w * w;
    const float theta = __builtin_amdgcn_sqrtf(t2);             // v_sqrt_f32 (== |w|)
    const float s     = __sinf(theta);                          // v_sin_f32 path
    const float c     = __cosf(theta);                          // v_cos_f32 path
    const float rTh   = __builtin_amdgcn_rcpf(theta + LIE_EPS); // v_rcp_f32
    const float rT2   = __builtin_amdgcn_rcpf(t2 + LIE_EPS);    // v_rcp_f32

    const float A  = s * rTh;                // sin(th)/th
    const float Bc = (1.0f - c) * rT2;       // (1-cos)/th^2
    const float C  = (1.0f - A) * rT2;       // (1-A)/th^2

    const float r00 = 1.0f - t2 * Bc;        // = r11
    const float wA  = w * A;                 // r10 = wA, r01 = -wA
    const float t00 = 1.0f - t2 * C;         // = t11
    const float wB  = w * Bc;                // t10 = wB, t01 = -wB

    const float tx = t00 * x - wB * y;
    const float ty = wB  * x + t00 * y;

    row0 = (v4f){r00, -wA, 0.0f, tx};
    row1 = (v4f){wA,  r00, 0.0f, ty};
}

__device__ __forceinline__ void emit4(v4f* __restrict__ o4, int k,
                                      float x, float y, float p) {
    v4f r0, r1;
    se2_rows(x, y, p, r0, r1);
    const v4f row2 = (v4f){0.0f, 0.0f, 1.0f, 0.0f};
    const v4f row3 = (v4f){0.0f, 0.0f, 0.0f, 1.0f};
    __builtin_nontemporal_store(r0,   o4 + 4 * k + 0);
    __builtin_nontemporal_store(r1,   o4 + 4 * k + 1);
    __builtin_nontemporal_store(row2, o4 + 4 * k + 2);
    __builtin_nontemporal_store(row3, o4 + 4 * k + 3);
}

__global__ __launch_bounds__(256)
void LieSE3_66649302499845_kernel(const float* __restrict__ uv,
                                  float* __restrict__ out,
                                  int nElem) {
    const int nGroups = nElem >> 2;                      // groups of 4 elements
    const int stride  = (int)(gridDim.x * blockDim.x);
    const int tid0    = (int)(blockIdx.x * blockDim.x + threadIdx.x);

    for (int g = tid0; g < nGroups; g += stride) {
        const v4f* __restrict__ in4 = (const v4f*)uv + (size_t)g * 3;

        // Prefetch this thread's NEXT tile into L2 (global_prefetch_b8).
        // Collectively: next ~12 MiB input slab staged in the 192 MB L2
        // while the current slab is consumed. Speculative -> OOB is safe.
        __builtin_prefetch((const void*)(in4 + (size_t)stride * 3), 0, 1);

        // 12 input floats for 4 elements: exactly 3x global_load_b128 (NT).
        const v4f a = __builtin_nontemporal_load(in4 + 0);
        const v4f b = __builtin_nontemporal_load(in4 + 1);
        const v4f c = __builtin_nontemporal_load(in4 + 2);

        // 4 matrices = 16x global_store_b128 (NT), contiguous 256 B / thread.
        v4f* __restrict__ o4 = (v4f*)out + (size_t)g * 16;
        emit4(o4, 0, a.x, a.y, a.z);
        emit4(o4, 1, a.w, b.x, b.y);
        emit4(o4, 2, b.z, b.w, c.x);
        emit4(o4, 3, c.y, c.z, c.w);
    }

    // Scalar tail for nElem % 4 != 0 (never taken for B = 4,194,304).
    const int rem = nElem & 3;
    if (tid0 < rem) {
        const int e = (nGroups << 2) + tid0;
        v4f r0, r1;
        se2_rows(uv[3 * e + 0], uv[3 * e + 1], uv[3 * e + 2], r0, r1);
        v4f* __restrict__ o4 = (v4f*)out + (size_t)e * 4;
        const v4f row2 = (v4f){0.0f, 0.0f, 1.0f, 0.0f};
        const v4f row3 = (v4f){0.0f, 0.0f, 0.0f, 1.0f};
        __builtin_nontemporal_store(r0,   o4 + 0);
        __builtin_nontemporal_store(r1,   o4 + 1);
        __builtin_nontemporal_store(row2, o4 + 2);
        __builtin_nontemporal_store(row3, o4 + 3);
    }
}

extern "C" void kernel_launch(void* const* d_in, const int* in_sizes, int n_in,
                              void* d_out, int out_size, void* d_ws, size_t ws_size,
                              hipStream_t stream) {
    (void)n_in; (void)out_size; (void)d_ws; (void)ws_size;
    const float* uv = (const float*)d_in[0];
    float* out      = (float*)d_out;
    const int nElem = in_sizes[0] / 3;                 // B = 4,194,304

    // Persistent grid: 1024 blocks x 8 wave32s = 8192 waves in flight,
    // ~4 tiles per thread so the one-tile-ahead prefetch has real targets.
    const int block = 256;
    int nGroups = nElem >> 2;
    int grid = (nGroups + block - 1) / block;
    if (grid > 1024) grid = 1024;
    if (grid < 1) grid = 1;
    LieSE3_66649302499845_kernel<<<grid, block, 0, stream>>>(uv, out, nElem);
}